// messageLayer_gla_71657234366630
// MI455X (gfx1250) — compile-verified
//
#include <hip/hip_runtime.h>
#include <math.h>

typedef __attribute__((ext_vector_type(2))) float v2f;
typedef __attribute__((ext_vector_type(8))) float v8f;

#define NB 2
#define HH 64
#define WW 64
#define LL 4096
#define DM 256
#define DF 128
#define NHH 8
#define DH 32
#define KC 32   // k-chunk staged in LDS

#if defined(__HIP_DEVICE_COMPILE__) && __has_builtin(__builtin_amdgcn_global_load_async_to_lds_b32)
#define HAVE_ASYNC_LDS 1
#else
#define HAVE_ASYNC_LDS 0
#endif

#if HAVE_ASYNC_LDS
#if __has_builtin(__builtin_amdgcn_s_wait_asynccnt)
#define WAITA(n) __builtin_amdgcn_s_wait_asynccnt(n)
#else
#define WAITA(n) asm volatile("s_wait_asynccnt %0" ::"i"(n) : "memory")
#endif
typedef __attribute__((address_space(1))) int* gint_p;
typedef __attribute__((address_space(3))) int* lint_p;
#else
#define WAITA(n)
#endif

__device__ __forceinline__ long bcol_map(int n, long Bsn, int ups)
{
  if (ups == 1) return (long)n * Bsn;
  const int y = n >> 6, x = n & 63;
  const int wp = WW / ups;
  return (long)((y / ups) * wp + (x / ups)) * Bsn;
}

// ---------------------------------------------------------------------------
// Simple 1-wave strided fp32 WMMA GEMM (used when M % 64 != 0)
// flags bit0 = relu on store, bit1 = accumulate (init acc from Y)
// ---------------------------------------------------------------------------
__global__ __launch_bounds__(32)
void wmma_gemm_f32(const float* __restrict__ A, const float* __restrict__ Bm,
                   float* __restrict__ Y,
                   long Asm, long Asc, long Asz,
                   long Bsc, long Bsn, long Bsz,
                   long Ysm, long Ysn, long Ysz,
                   int C, int flags, int ups)
{
  const int lane = threadIdx.x;
  const int mrow = lane & 15;
  const int half = lane >> 4;
  const int m0 = blockIdx.y * 16;
  const int n0 = blockIdx.x * 16;
  const int z  = blockIdx.z;
  const float* Az = A + (long)z * Asz;
  const float* Bz = Bm + (long)z * Bsz;
  float* Yz = Y + (long)z * Ysz;
  const int n = n0 + mrow;
  const long bcol = bcol_map(n, Bsn, ups);
  v8f acc = {0.f,0.f,0.f,0.f,0.f,0.f,0.f,0.f};
  if (flags & 2) {
#pragma unroll
    for (int r = 0; r < 8; ++r)
      acc[r] = Yz[(long)(m0 + r + 8*half) * Ysm + (long)n * Ysn];
  }
  const long arow = (long)(m0 + mrow) * Asm;
  for (int c = 0; c < C; c += 4) {
    const int ca = c + 2*half;
    v2f av, bv;
    av[0] = Az[arow + (long)ca * Asc];
    av[1] = Az[arow + (long)(ca+1) * Asc];
    bv[0] = Bz[(long)ca * Bsc + bcol];
    bv[1] = Bz[(long)(ca+1) * Bsc + bcol];
    acc = __builtin_amdgcn_wmma_f32_16x16x4_f32(false, av, false, bv,
                                                (short)0, acc, false, false);
  }
#pragma unroll
  for (int r = 0; r < 8; ++r) {
    float v = acc[r];
    if (flags & 1) v = fmaxf(v, 0.f);
    Yz[(long)(m0 + r + 8*half) * Ysm + (long)n * Ysn] = v;
  }
}

// ---------------------------------------------------------------------------
// 4-wave staged WMMA GEMM: M-tile 64, shared B tile (KCx16) double-buffered
// in LDS via async global->LDS DMA (ASYNCcnt pipelined). Requires M%64==0,
// C%KC==0, N%16==0.
// ---------------------------------------------------------------------------
__global__ __launch_bounds__(128)
void wmma_gemm_staged(const float* __restrict__ A, const float* __restrict__ Bm,
                      float* __restrict__ Y,
                      long Asm, long Asc, long Asz,
                      long Bsc, long Bsn, long Bsz,
                      long Ysm, long Ysn, long Ysz,
                      int C, int flags, int ups)
{
  __shared__ float bs[2][16][KC + 2];   // [buf][col][row], pad keeps banks spread

  const int t    = threadIdx.x;
  const int lane = t & 31;
  const int w    = t >> 5;            // wave id (0..3)
  const int mrow = lane & 15;
  const int half = lane >> 4;
  const int m0 = blockIdx.y * 64 + w * 16;
  const int n0 = blockIdx.x * 16;
  const int z  = blockIdx.z;
  const float* Az = A + (long)z * Asz;
  const float* Bz = Bm + (long)z * Bsz;
  float* Yz = Y + (long)z * Ysz;

  // writer mapping: col fixed per thread, rows strided so each async op
  // covers 16 contiguous columns x 2 rows (coalesced)
  const int colw = t & 15;
  const int rowb = t >> 4;            // 0..7
  const long bcol_w = bcol_map(n0 + colw, Bsn, ups);

  const int n = n0 + mrow;
  v8f acc = {0.f,0.f,0.f,0.f,0.f,0.f,0.f,0.f};
  if (flags & 2) {
#pragma unroll
    for (int r = 0; r < 8; ++r)
      acc[r] = Yz[(long)(m0 + r + 8*half) * Ysm + (long)n * Ysn];
  }
  const long arow = (long)(m0 + mrow) * Asm;

  const int nch = C / KC;

  // stage chunk 0
#pragma unroll
  for (int i = 0; i < 4; ++i) {
    const int row = rowb + 8 * i;
    const long g = (long)row * Bsc + bcol_w;
#if HAVE_ASYNC_LDS
    __builtin_amdgcn_global_load_async_to_lds_b32(
        (gint_p)(Bz + g), (lint_p)&bs[0][colw][row], 0, 0);
#else
    bs[0][colw][row] = Bz[g];
#endif
  }

  for (int ch = 0; ch < nch; ++ch) {
    const int nxt = ch + 1;
    if (nxt < nch) {
      const int cb2 = nxt * KC;
#pragma unroll
      for (int i = 0; i < 4; ++i) {
        const int row = rowb + 8 * i;
        const long g = (long)(cb2 + row) * Bsc + bcol_w;
#if HAVE_ASYNC_LDS
        __builtin_amdgcn_global_load_async_to_lds_b32(
            (gint_p)(Bz + g), (lint_p)&bs[nxt & 1][colw][row], 0, 0);
#else
        bs[nxt & 1][colw][row] = Bz[g];
#endif
      }
      // prefetch next A (weight) chunk
      __builtin_prefetch(&Az[arow + (long)(nxt * KC) * Asc], 0, 1);
    }
#if HAVE_ASYNC_LDS
    if (nxt < nch) { WAITA(4); } else { WAITA(0); }
#endif
    __syncthreads();
    const int cb = ch * KC;
    const float* bb = &bs[ch & 1][mrow][0];
#pragma unroll
    for (int kk = 0; kk < KC; kk += 4) {
      const int ca = cb + kk + 2 * half;
      const int lr = kk + 2 * half;
      v2f av, bv;
      av[0] = Az[arow + (long)ca * Asc];
      av[1] = Az[arow + (long)(ca + 1) * Asc];
      bv[0] = bb[lr];
      bv[1] = bb[lr + 1];
      acc = __builtin_amdgcn_wmma_f32_16x16x4_f32(false, av, false, bv,
                                                  (short)0, acc, false, false);
    }
    __syncthreads();
  }

#pragma unroll
  for (int r = 0; r < 8; ++r) {
    float v = acc[r];
    if (flags & 1) v = fmaxf(v, 0.f);
    Yz[(long)(m0 + r + 8*half) * Ysm + (long)n * Ysn] = v;
  }
}

// ---------------------------------------------------------------------------
// 4-wave staged 3x3 conv via WMMA: OC-tile 64 x pixel-tile 16 (one image row
// segment). X halo (3 rows x 18 cols x KC chans) double-buffered in LDS via
// async DMA; border zero-padding applied at consumption with per-lane selects
// so EXEC stays full and per-wave ASYNCcnt is uniform (24 issues/chunk).
// W (OC,IC,3,3), X (NB,IC,64,64)
// ---------------------------------------------------------------------------
__global__ __launch_bounds__(128)
void conv3_wmma_staged(const float* __restrict__ Wt, const float* __restrict__ X,
                       float* __restrict__ Y, int OC, int IC)
{
  __shared__ float xs[2][3][32][KC + 2];  // [buf][row][col][chan]

  const int t    = threadIdx.x;
  const int lane = t & 31;
  const int w    = t >> 5;
  const int mrow = lane & 15;
  const int half = lane >> 4;
  const int n0  = blockIdx.x * 16;
  const int oc0 = blockIdx.y * 64 + w * 16;
  const int b   = blockIdx.z;
  const int y   = n0 >> 6;       // uniform row for this pixel tile
  const int x0  = n0 & 63;
  const float* Xb = X + (long)b * IC * LL;

  // writer: lane -> column (contiguous x, clamped at borders)
  const int xw  = x0 + lane - 1;
  const int xwc = xw < 0 ? 0 : (xw > WW - 1 ? WW - 1 : xw);
  const int nch = IC / KC;

  v8f acc = {0.f,0.f,0.f,0.f,0.f,0.f,0.f,0.f};

  // stage chunk 0: 24 iterations/wave -> (row, chan) pairs; col = lane
#pragma unroll
  for (int it = 0; it < 24; ++it) {
    const int idx2 = w + 4 * it;        // 0..95
    const int row  = idx2 >> 5;         // 0..2
    const int chan = idx2 & 31;
    const int yy   = y + row - 1;
    const int yyc  = yy < 0 ? 0 : (yy > HH - 1 ? HH - 1 : yy);
    const long g = (long)chan * LL + (long)yyc * WW + xwc;
#if HAVE_ASYNC_LDS
    __builtin_amdgcn_global_load_async_to_lds_b32(
        (gint_p)(Xb + g), (lint_p)&xs[0][row][lane][chan], 0, 0);
#else
    xs[0][row][lane][chan] = Xb[g];
#endif
  }

  for (int ch = 0; ch < nch; ++ch) {
    const int nxt = ch + 1;
    if (nxt < nch) {
      const int cb2 = nxt * KC;
#pragma unroll
      for (int it = 0; it < 24; ++it) {
        const int idx2 = w + 4 * it;
        const int row  = idx2 >> 5;
        const int chan = idx2 & 31;
        const int yy   = y + row - 1;
        const int yyc  = yy < 0 ? 0 : (yy > HH - 1 ? HH - 1 : yy);
        const long g = (long)(cb2 + chan) * LL + (long)yyc * WW + xwc;
#if HAVE_ASYNC_LDS
        __builtin_amdgcn_global_load_async_to_lds_b32(
            (gint_p)(Xb + g), (lint_p)&xs[nxt & 1][row][lane][chan], 0, 0);
#else
        xs[nxt & 1][row][lane][chan] = Xb[g];
#endif
      }
    }
#if HAVE_ASYNC_LDS
    if (nxt < nch) { WAITA(24); } else { WAITA(0); }
#endif
    __syncthreads();
    const int cb = ch * KC;
    for (int dy = 0; dy < 3; ++dy) {
      const int yy = y + dy - 1;
      const bool vy = (yy >= 0) && (yy < HH);
      for (int dx = 0; dx < 3; ++dx) {
        const int xx = x0 + mrow + dx - 1;
        const bool v = vy && (xx >= 0) && (xx < WW);
        const int tap = dy * 3 + dx;
        const long wbase = (long)(oc0 + mrow) * IC * 9 + tap;
        const float* bb = &xs[ch & 1][dy][mrow + dx][0];
#pragma unroll
        for (int kk = 0; kk < KC; kk += 4) {
          const int ca = kk + 2 * half;
          v2f av, bv;
          av[0] = Wt[wbase + (long)(cb + ca) * 9];
          av[1] = Wt[wbase + (long)(cb + ca + 1) * 9];
          bv[0] = v ? bb[ca] : 0.f;
          bv[1] = v ? bb[ca + 1] : 0.f;
          acc = __builtin_amdgcn_wmma_f32_16x16x4_f32(false, av, false, bv,
                                                      (short)0, acc, false, false);
        }
      }
    }
    __syncthreads();
  }

  float* Yb = Y + (long)b * OC * LL;
#pragma unroll
  for (int r = 0; r < 8; ++r)
    Yb[(long)(oc0 + r + 8*half) * LL + n0 + mrow] = acc[r];
}

// ---------------------------------------------------------------------------
// avg pool (NB,C,64,64)->(NB,C,64/s,64/s); total = NB*C*(64/s)^2
// ---------------------------------------------------------------------------
__global__ void avgpool_k(const float* __restrict__ X, float* __restrict__ Y,
                          int s, long total)
{
  const int Ho = HH / s;
  const float inv = 1.f / (float)(s * s);
  for (long i = (long)blockIdx.x * 256 + threadIdx.x; i < total;
       i += (long)gridDim.x * 256) {
    int xo = (int)(i % Ho); long r = i / Ho;
    int yo = (int)(r % Ho); long bc = r / Ho;
    const float* xp = X + (bc * HH + (long)yo * s) * WW + (long)xo * s;
    float a = 0.f;
    for (int wy = 0; wy < s; ++wy)
      for (int wx = 0; wx < s; ++wx)
        a += xp[(long)wy * WW + wx];
    Y[i] = a * inv;
  }
}

__global__ void minpool_k(const float* __restrict__ X, float* __restrict__ Y,
                          int s, long total)
{
  const int Ho = HH / s;
  for (long i = (long)blockIdx.x * 256 + threadIdx.x; i < total;
       i += (long)gridDim.x * 256) {
    int xo = (int)(i % Ho); long r = i / Ho;
    int yo = (int)(r % Ho); long b = r / Ho;
    const float* xp = X + (b * HH + (long)yo * s) * WW + (long)xo * s;
    float mn = 3.4e38f;
    for (int wy = 0; wy < s; ++wy)
      for (int wx = 0; wx < s; ++wx)
        mn = fminf(mn, xp[(long)wy * WW + wx]);
    Y[i] = mn;
  }
}

// ---------------------------------------------------------------------------
// level-0 masked softmax over s=256; att (z=B*NH, 256 l, 256 s)
// ---------------------------------------------------------------------------
__global__ __launch_bounds__(256)
void softmax0_k(float* __restrict__ att, const float* __restrict__ m0p,
                const float* __restrict__ m1p, const float* __restrict__ temp)
{
  __shared__ float red[256];
  const int l = blockIdx.x;
  const int z = blockIdx.y;
  const int b = z >> 3;
  const int t = threadIdx.x;
  const float scale = temp[0] * 0.17677669529663687f;  // temp / sqrt(32)
  float* row = att + ((long)z * 256 + l) * 256;
  const bool mk0 = (m0p[b * 256 + l] != 0.f);
  const float v = row[t];
  const bool m = mk0 && (m1p[b * 256 + t] != 0.f);
  const float lg = (m ? v : -1e8f) * scale;
  red[t] = lg; __syncthreads();
  for (int o = 128; o; o >>= 1) { if (t < o) red[t] = fmaxf(red[t], red[t + o]); __syncthreads(); }
  const float mx = red[0]; __syncthreads();
  const float e = expf(lg - mx);
  red[t] = e; __syncthreads();
  for (int o = 128; o; o >>= 1) { if (t < o) red[t] += red[t + o]; __syncthreads(); }
  row[t] = e / red[0];
}

// ---------------------------------------------------------------------------
// Group attention for levels 1,2: one workgroup per (b, group)
// ---------------------------------------------------------------------------
__global__ __launch_bounds__(256)
void group_attn_k(const float* __restrict__ Q, const float* __restrict__ K,
                  const float* __restrict__ V, const float* __restrict__ M0,
                  const float* __restrict__ OFF, const float* __restrict__ SPAN,
                  float* __restrict__ OUT,
                  int Hq, int Wq, int Hk, int Wk, int Gw)
{
  const int G  = (Hq >> 1) * (Wq >> 1);
  const int b  = blockIdx.x / G;
  const int g  = blockIdx.x % G;
  const int gy = g / Gw;
  const int gx = g % Gw;
  const int t  = threadIdx.x;
  const int HWq = Hq * Wq;
  const int HWk = Hk * Wk;

  __shared__ float q_s[DM][4];
  __shared__ float kg_s[DH][64];
  __shared__ float vg_s[DH][64];
  __shared__ float logit[4][64];
  __shared__ float cw[64][4];
  __shared__ int   cidxs[64][4];
  __shared__ float msk[64];

  if (t < 64) {
    const int m = t;
    const int i = m >> 3, j = m & 7;
    const long ob = (long)(b * G + g) * 2;
    const float sx = OFF[ob]     + ((float)i - 3.5f) * SPAN[ob];
    const float sy = OFF[ob + 1] + ((float)j - 3.5f) * SPAN[ob + 1];
    const float gxf = sx * (float)(Wk - 1) / (float)Wk;
    const float gyf = sy * (float)(Hk - 1) / (float)Hk;
    const float x0f = floorf(gxf), y0f = floorf(gyf);
    const float wx = gxf - x0f, wy = gyf - y0f;
    const float cxs[4] = {x0f, x0f + 1.f, x0f, x0f + 1.f};
    const float cys[4] = {y0f, y0f, y0f + 1.f, y0f + 1.f};
    const float cws[4] = {(1.f-wx)*(1.f-wy), wx*(1.f-wy), (1.f-wx)*wy, wx*wy};
#pragma unroll
    for (int cc = 0; cc < 4; ++cc) {
      const float xi = cxs[cc], yi = cys[cc];
      const bool valid = (xi >= 0.f) && (xi <= (float)(Wk - 1)) &&
                         (yi >= 0.f) && (yi <= (float)(Hk - 1));
      const int xc = (int)fminf(fmaxf(xi, 0.f), (float)(Wk - 1));
      const int yc = (int)fminf(fmaxf(yi, 0.f), (float)(Hk - 1));
      cidxs[m][cc] = yc * Wk + xc;
      cw[m][cc] = valid ? cws[cc] : 0.f;
    }
    const float xr = rintf(sx - 0.5f), yr = rintf(sy - 0.5f);
    const bool nv = (xr >= 0.f) && (xr <= (float)(Wk - 1)) &&
                    (yr >= 0.f) && (yr <= (float)(Hk - 1));
    const int xc = (int)fminf(fmaxf(xr, 0.f), (float)(Wk - 1));
    const int yc = (int)fminf(fmaxf(yr, 0.f), (float)(Hk - 1));
    const float mv = nv ? M0[(long)b * HWk + yc * Wk + xc] : 0.f;
    msk[m] = (mv == 1.f) ? 1.f : 0.f;
  }
  for (int idx = t; idx < DM * 4; idx += 256) {
    const int c = idx >> 2, n = idx & 3;
    const int y = gy * 2 + (n >> 1), x = gx * 2 + (n & 1);
    q_s[c][n] = Q[(long)(b * DM + c) * HWq + (long)y * Wq + x];
  }
  __syncthreads();

  const float rs32 = 0.17677669529663687f;
  for (int h = 0; h < NHH; ++h) {
    for (int idx = t; idx < DH * 64; idx += 256) {
      const int c = idx >> 6, m = idx & 63;
      const long base = (long)(b * DM + h * DH + c) * HWk;
      float kv = 0.f, vv = 0.f;
#pragma unroll
      for (int cc = 0; cc < 4; ++cc) {
        const float w = cw[m][cc];
        const int ci = cidxs[m][cc];
        kv += w * K[base + ci];
        vv += w * V[base + ci];
      }
      kg_s[c][m] = kv;
      vg_s[c][m] = vv;
    }
    __syncthreads();
    {
      const int n = t >> 6, m = t & 63;
      float dot = 0.f;
#pragma unroll
      for (int d = 0; d < DH; ++d)
        dot += q_s[h * DH + d][n] * kg_s[d][m];
      logit[n][m] = (msk[m] != 0.f ? dot : -1e8f) * rs32;
    }
    __syncthreads();
    if (t < 4) {
      float mx = -3.4e38f;
      for (int m = 0; m < 64; ++m) mx = fmaxf(mx, logit[t][m]);
      float s = 0.f;
      for (int m = 0; m < 64; ++m) { float e = expf(logit[t][m] - mx); logit[t][m] = e; s += e; }
      const float inv = 1.f / s;
      for (int m = 0; m < 64; ++m) logit[t][m] *= inv;
    }
    __syncthreads();
    if (t < 128) {
      const int d = t >> 2, n = t & 3;
      float s = 0.f;
#pragma unroll
      for (int m = 0; m < 64; ++m)
        s += logit[n][m] * vg_s[d][m];
      // reference reshapes (g,n) flat directly into row-major image
      OUT[(long)(b * DM + h * DH + d) * HWq + (long)g * 4 + n] = s;
    }
    __syncthreads();
  }
}

// ---------------------------------------------------------------------------
// channel LayerNorm (ddof=1), one wave per pixel. Safe in-place.
// ---------------------------------------------------------------------------
__global__ __launch_bounds__(256)
void ln2d_k(const float* __restrict__ X, float* __restrict__ Y,
            const float* __restrict__ gam, const float* __restrict__ bet, int C)
{
  const int wid = (blockIdx.x * 256 + threadIdx.x) >> 5;
  const int lane = threadIdx.x & 31;
  if (wid >= NB * LL) return;
  const int b = wid / LL, l = wid % LL;
  const float* xp = X + (long)b * C * LL + l;
  float s = 0.f, ss = 0.f;
  for (int c = lane; c < C; c += 32) {
    const float v = xp[(long)c * LL];
    s += v; ss += v * v;
  }
  for (int o = 16; o; o >>= 1) { s += __shfl_xor(s, o); ss += __shfl_xor(ss, o); }
  const float mean = s / (float)C;
  const float var = (ss - (float)C * mean * mean) / (float)(C - 1);
  const float inv = 1.f / (sqrtf(fmaxf(var, 0.f)) + 1e-6f);
  float* yp = Y + (long)b * C * LL + l;
  for (int c = lane; c < C; c += 32) {
    const float v = xp[(long)c * LL];
    yp[(long)c * LL] = gam[c] * (v - mean) * inv + bet[c];
  }
}

// ---------------------------------------------------------------------------
// flow head: 4x64 GEMM + sigmoid scaling -> (b,H,W,4)
// ---------------------------------------------------------------------------
__global__ void flow_head_k(const float* __restrict__ Hf,
                            const float* __restrict__ W2,
                            float* __restrict__ Fout)
{
  const int i = blockIdx.x * 256 + threadIdx.x;
  if (i >= NB * LL) return;
  const int b = i / LL, n = i % LL;
  float a0 = 0.f, a1 = 0.f, a2 = 0.f, a3 = 0.f;
  const float* hp = Hf + (long)b * 64 * LL + n;
  for (int c = 0; c < 64; ++c) {
    const float hv = hp[(long)c * LL];
    a0 += W2[c] * hv;
    a1 += W2[64 + c] * hv;
    a2 += W2[128 + c] * hv;
    a3 += W2[192 + c] * hv;
  }
  float* fp = Fout + (long)i * 4;
  fp[0] = 64.f / (1.f + expf(-a0));
  fp[1] = 64.f / (1.f + expf(-a1));
  fp[2] = a2;
  fp[3] = a3;
}

// ---------------------------------------------------------------------------
// offset/span pooling from flow (b,H,W,4); OFF,SPAN (b,Gd,Gd,2)
// ---------------------------------------------------------------------------
__global__ void offspan_k(const float* __restrict__ F, float* __restrict__ OFF,
                          float* __restrict__ SPAN, int Gd, int k, float sdiv)
{
  const int i = blockIdx.x * 256 + threadIdx.x;
  const int total = NB * Gd * Gd * 2;
  if (i >= total) return;
  const int c = i & 1; const int g = i >> 1;
  const int gx = g % Gd; const int gy = (g / Gd) % Gd; const int b = g / (Gd * Gd);
  float so = 0.f, sp = 0.f;
  for (int wy = 0; wy < k; ++wy)
    for (int wx = 0; wx < k; ++wx) {
      const int y = gy * k + wy, x = gx * k + wx;
      const float* f = F + (((long)b * HH + y) * WW + x) * 4;
      so += f[c];
      const float var = expf(0.5f * f[2 + c]) * 5.f;   // RS = 5
      sp += fmaxf(var * 0.25f, 1.f);                   // *2/NS1
    }
  const float inv = 1.f / (float)(k * k);
  OFF[i] = so * inv / sdiv;
  SPAN[i] = sp * inv;
}

// ---------------------------------------------------------------------------
// residual add + split to x / flow-feature outputs
// ---------------------------------------------------------------------------
__global__ void final_add_k(const float* __restrict__ Xin,
                            const float* __restrict__ FFin,
                            const float* __restrict__ C3,
                            float* __restrict__ Xout, float* __restrict__ FFout)
{
  const long total = (long)NB * (DM + DF) * LL;
  for (long i = (long)blockIdx.x * 256 + threadIdx.x; i < total;
       i += (long)gridDim.x * 256) {
    const int l = (int)(i % LL); long r = i / LL;
    const int c = (int)(r % (DM + DF)); const int b = (int)(r / (DM + DF));
    const float add = C3[i];
    if (c < DM)
      Xout[((long)b * DM + c) * LL + l] = Xin[((long)b * DM + c) * LL + l] + add;
    else
      FFout[((long)b * DF + (c - DM)) * LL + l] =
          FFin[((long)b * DF + (c - DM)) * LL + l] + add;
  }
}

// ---------------------------------------------------------------------------
// host side
// ---------------------------------------------------------------------------
static void launch_gemm(hipStream_t st, const float* A, const float* B, float* Y,
                        long Asm, long Asc, long Asz,
                        long Bsc, long Bsn, long Bsz,
                        long Ysm, long Ysn, long Ysz,
                        int M, int N, int C, int Z, int flags, int ups)
{
  if ((M % 64) == 0 && (C % KC) == 0) {
    dim3 g((unsigned)(N / 16), (unsigned)(M / 64), (unsigned)Z), bl(128, 1, 1);
    wmma_gemm_staged<<<g, bl, 0, st>>>(A, B, Y, Asm, Asc, Asz, Bsc, Bsn, Bsz,
                                       Ysm, Ysn, Ysz, C, flags, ups);
  } else {
    dim3 g((unsigned)(N / 16), (unsigned)(M / 16), (unsigned)Z), bl(32, 1, 1);
    wmma_gemm_f32<<<g, bl, 0, st>>>(A, B, Y, Asm, Asc, Asz, Bsc, Bsn, Bsz,
                                    Ysm, Ysn, Ysz, C, flags, ups);
  }
}

extern "C" void kernel_launch(void* const* d_in, const int* in_sizes, int n_in,
                              void* d_out, int out_size, void* d_ws, size_t ws_size,
                              hipStream_t stream)
{
  (void)in_sizes; (void)n_in; (void)out_size; (void)ws_size;
  const float* x0   = (const float*)d_in[0];
  const float* x1   = (const float*)d_in[1];
  const float* ff0  = (const float*)d_in[2];
  const float* ff1  = (const float*)d_in[3];
  const float* pos0 = (const float*)d_in[4];
  const float* pos1 = (const float*)d_in[5];
  const float* mask0 = (const float*)d_in[6];
  const float* mask1 = (const float*)d_in[7];
  const float* w_fd1 = (const float*)d_in[8];
  const float* w_fd2 = (const float*)d_in[9];
  const float* w_q  = (const float*)d_in[10];
  const float* w_k  = (const float*)d_in[11];
  const float* w_v  = (const float*)d_in[12];
  const float* w_mh1 = (const float*)d_in[13];
  const float* w_mh2 = (const float*)d_in[14];
  const float* temp  = (const float*)d_in[15];
  const float* w_mf1 = (const float*)d_in[16];
  const float* w_mf2 = (const float*)d_in[17];
  const float* g1 = (const float*)d_in[18];
  const float* b1 = (const float*)d_in[19];
  const float* g2 = (const float*)d_in[20];
  const float* b2 = (const float*)d_in[21];
  float* dout = (float*)d_out;
  float* ws   = (float*)d_ws;
  hipStream_t st = stream;

  // d_out layout (floats)
  const size_t X0N = 0, X1N = 2097152, FF0N = 4194304, FF1N = 5242880,
               FLW0 = 6291456, FLW1 = 6324224;
  // workspace layout (floats)
  float* Q   = ws + 0;
  float* Kb  = ws + 2097152;
  float* Vb  = ws + 4194304;
  float* Q4  = ws + 6291456;
  float* K4  = ws + 6422528;
  float* V4  = ws + 6553600;
  float* Q2  = ws + 6684672;
  float* K2  = ws + 7208960;
  float* V2  = ws + 7733248;
  float* ATT = ws + 8257536;
  float* MSG0  = ws + 9306112;
  float* MSG1  = ws + 9437184;
  float* MSG2L = ws + 9961472;
  float* H1    = ws + 12058624;
  float* MSGF  = ws + 14155776;
  float* LNB   = ws + 16252928;
  float* HMF   = ws + 18350080;
  float* C3    = ws + 21495808;
  float* FLOWH = ws + 24641536;
  float* OFF1  = ws + 25165824;
  float* SPAN1 = ws + 25166848;
  float* OFF2  = ws + 25167872;
  float* SPAN2 = ws + 25171968;
  float* M04   = ws + 25176064;
  float* M14   = ws + 25176576;
  float* M02   = ws + 25177088;

  // ---- decode_flow for both directions ----
  launch_gemm(st, w_fd1, ff0, FLOWH, 128, 1, 0, LL, 1, (long)DF * LL,
              LL, 1, 64L * LL, 64, LL, DF, NB, 1, 1);
  flow_head_k<<<(NB * LL + 255) / 256, 256, 0, st>>>(FLOWH, w_fd2, dout + FLW0);
  launch_gemm(st, w_fd1, ff1, FLOWH, 128, 1, 0, LL, 1, (long)DF * LL,
              LL, 1, 64L * LL, 64, LL, DF, NB, 1, 1);
  flow_head_k<<<(NB * LL + 255) / 256, 256, 0, st>>>(FLOWH, w_fd2, dout + FLW1);

  for (int dir = 0; dir < 2; ++dir) {
    const float* xa   = dir ? x1 : x0;
    const float* xb   = dir ? x0 : x1;
    const float* ffa  = dir ? ff1 : ff0;
    const float* posb = dir ? pos0 : pos1;
    const float* m0p  = dir ? mask1 : mask0;
    const float* m1p  = dir ? mask0 : mask1;
    float* xout  = dout + (dir ? X1N : X0N);
    float* ffout = dout + (dir ? FF1N : FF0N);
    const float* flowp = dout + (dir ? FLW1 : FLW0);

    const long SB = (long)DM * LL;     // 1048576
    // q/k/v projections
    launch_gemm(st, w_q, xa, Q,  256, 1, 0, LL, 1, SB, LL, 1, SB, DM, LL, DM, NB, 0, 1);
    launch_gemm(st, w_k, xb, Kb, 256, 1, 0, LL, 1, SB, LL, 1, SB, DM, LL, DM, NB, 0, 1);
    launch_gemm(st, w_v, xb, Vb, 320, 1, 0, LL, 1, SB, LL, 1, SB, DM, LL, DM, NB, 0, 1);
    launch_gemm(st, w_v + 256, posb, Vb, 320, 1, 0, LL, 1, 64L * LL,
                LL, 1, SB, DM, LL, 64, NB, 2, 1);

    // pyramid pools
    avgpool_k<<<512, 256, 0, st>>>(Q,  Q4, 4, (long)NB * DM * 256);
    avgpool_k<<<512, 256, 0, st>>>(Kb, K4, 4, (long)NB * DM * 256);
    avgpool_k<<<512, 256, 0, st>>>(Vb, V4, 4, (long)NB * DM * 256);
    avgpool_k<<<2048, 256, 0, st>>>(Q,  Q2, 2, (long)NB * DM * 1024);
    avgpool_k<<<2048, 256, 0, st>>>(Kb, K2, 2, (long)NB * DM * 1024);
    avgpool_k<<<2048, 256, 0, st>>>(Vb, V2, 2, (long)NB * DM * 1024);
    minpool_k<<<2, 256, 0, st>>>(m0p, M04, 4, (long)NB * 256);
    minpool_k<<<2, 256, 0, st>>>(m1p, M14, 4, (long)NB * 256);
    minpool_k<<<8, 256, 0, st>>>(m0p, M02, 2, (long)NB * 1024);

    // level-0 full attention: logits = Q^T K per (b,h); z stride = DH*256
    launch_gemm(st, Q4, K4, ATT, 1, 256, 8192, 256, 1, 8192,
                256, 1, 65536, 256, 256, DH, NB * NHH, 0, 1);
    softmax0_k<<<dim3(256, NB * NHH), 256, 0, st>>>(ATT, M04, M14, temp);
    launch_gemm(st, V4, ATT, MSG0, 256, 1, 8192, 1, 256, 65536,
                256, 1, 8192, DH, 256, 256, NB * NHH, 0, 1);

    // flow-derived sampling offsets / spans
    offspan_k<<<(NB * 16 * 16 * 2 + 255) / 256, 256, 0, st>>>(flowp, OFF1, SPAN1, 16, 4, 2.f);
    offspan_k<<<(NB * 32 * 32 * 2 + 255) / 256, 256, 0, st>>>(flowp, OFF2, SPAN2, 32, 2, 1.f);

    // group attention levels 1 & 2
    group_attn_k<<<NB * 256, 256, 0, st>>>(Q2, K2, V2, M02, OFF1, SPAN1, MSG1,
                                           32, 32, 32, 32, 16);
    group_attn_k<<<NB * 1024, 256, 0, st>>>(Q, Kb, Vb, m0p, OFF2, SPAN2, MSG2L,
                                            64, 64, 64, 64, 32);

    // mh1: fused concat+upsample via split-column accumulating GEMMs
    launch_gemm(st, w_mh1,       MSG0,  H1, 768, 1, 0, 256, 1, 65536,
                LL, 1, SB, DM, LL, DM, NB, 0, 4);
    launch_gemm(st, w_mh1 + 256, MSG1,  H1, 768, 1, 0, 1024, 1, 262144,
                LL, 1, SB, DM, LL, DM, NB, 2, 2);
    launch_gemm(st, w_mh1 + 512, MSG2L, H1, 768, 1, 0, LL, 1, SB,
                LL, 1, SB, DM, LL, DM, NB, 3, 1);
    launch_gemm(st, w_mh2, H1, MSGF, 256, 1, 0, LL, 1, SB,
                LL, 1, SB, DM, LL, DM, NB, 0, 1);

    ln2d_k<<<1024, 256, 0, st>>>(MSGF, LNB, g1, b1, DM);

    // mf1: [x, ff, ln(msg)] -> 384, relu (split-column accumulation)
    const long SH = (long)(DM + DF) * LL;   // 1572864
    launch_gemm(st, w_mf1,       xa,  HMF, 640, 1, 0, LL, 1, SB,
                LL, 1, SH, DM + DF, LL, DM, NB, 0, 1);
    launch_gemm(st, w_mf1 + 256, ffa, HMF, 640, 1, 0, LL, 1, (long)DF * LL,
                LL, 1, SH, DM + DF, LL, DF, NB, 2, 1);
    launch_gemm(st, w_mf1 + 384, LNB, HMF, 640, 1, 0, LL, 1, SB,
                LL, 1, SH, DM + DF, LL, DM, NB, 3, 1);

    // conv3x3 (staged async/LDS) + LN + residual
    conv3_wmma_staged<<<dim3(256, (DM + DF) / 64, NB), 128, 0, st>>>(
        w_mf2, HMF, C3, DM + DF, DM + DF);
    ln2d_k<<<1024, 256, 0, st>>>(C3, C3, g2, b2, DM + DF);
    final_add_k<<<4096, 256, 0, st>>>(xa, ffa, C3, xout, ffout);
  }
}